// MultiHeadAttention_74998718923417
// MI455X (gfx1250) — compile-verified
//
#include <hip/hip_runtime.h>
#include <hip/hip_bf16.h>
#include <math.h>

// Problem constants (match reference)
#define B_ 4
#define T_ 1024
#define C_ 1024
#define H_ 16
#define D_ 64          // head dim
#define C3_ (3 * C_)   // 3072

typedef __attribute__((ext_vector_type(2))) float v2f;
typedef __attribute__((ext_vector_type(8))) float v8f;
typedef __attribute__((ext_vector_type(4))) int v4i;

// ---------------------------------------------------------------------------
// Async global->LDS (CDNA5 GLOBAL_LOAD_ASYNC_TO_LDS_*, ASYNCcnt-tracked).
// Probe-derived signatures: param0 = AS1 (global/__device__) pointer to the
// element type (v4i for b128, int for b32), non-const; param1 = LDS pointer.
// Guarded: falls back to plain load/store pairs if builtins are absent.
// ---------------------------------------------------------------------------
#if __has_builtin(__builtin_amdgcn_global_load_async_to_lds_b128) && \
    __has_builtin(__builtin_amdgcn_global_load_async_to_lds_b32)
#define HAVE_ASYNC_G2L 1
#else
#define HAVE_ASYNC_G2L 0
#endif

typedef __attribute__((address_space(1))) v4i  gv4i;
typedef __attribute__((address_space(1))) int  gint;
typedef __attribute__((address_space(3))) v4i  lv4i;
typedef __attribute__((address_space(3))) int  lint;

__device__ __forceinline__ void g2l_b128(const float* g, float* l) {
#if HAVE_ASYNC_G2L
  __builtin_amdgcn_global_load_async_to_lds_b128((gv4i*)g, (lv4i*)l, 0, 0);
#else
  *reinterpret_cast<float4*>(l) = *reinterpret_cast<const float4*>(g);
#endif
}

__device__ __forceinline__ void g2l_b32(const float* g, float* l) {
#if HAVE_ASYNC_G2L
  __builtin_amdgcn_global_load_async_to_lds_b32((gint*)g, (lint*)l, 0, 0);
#else
  *l = *g;
#endif
}

__device__ __forceinline__ void stage_wait() {
#if HAVE_ASYNC_G2L
#if __has_builtin(__builtin_amdgcn_s_wait_asynccnt)
  __builtin_amdgcn_s_wait_asynccnt(0);
#else
  asm volatile("s_wait_asynccnt 0x0" ::: "memory");
#endif
#endif
}

// ---------------------------------------------------------------------------
// fp32 WMMA GEMM with bias:  Cmat[M,N] = A[M,K] * Bm[K,N] + bias[N]
// Block = 128 threads (4 waves). Block tile 64x64, K-tile 16, double-buffered
// LDS: tile k+1 streams in (async DMA when available) while WMMAs consume
// tile k. B tile stored TRANSPOSED in LDS so each B-frag is one aligned
// ds_load_b64. Row pads of 20 floats keep b64 frag loads bank-conflict-free
// (20n mod 64 injective over [0,16), 0-mod-4 class; +2 half-wave offset fills
// the 2-mod-4 class) and rows stay 16B-aligned (80B) for b128 staging.
// ---------------------------------------------------------------------------
#define TM 64
#define TN 64
#define TK 16
#define APAD 20
#define BPAD 20

__device__ __forceinline__ void stage_gemm_tile(float (*As)[APAD], float (*Bst)[BPAD],
                                                const float* __restrict__ Ag,  // A + bm*K + k0
                                                const float* __restrict__ Bg,  // Bm + k0*N + bn
                                                unsigned K, unsigned N, int tid) {
#if HAVE_ASYNC_G2L
  // A tile 64x16: 256 16B chunks, 2 per thread (b128 DMA, no VGPR round trip)
#pragma unroll
  for (int j = 0; j < 2; ++j) {
    int i = tid + j * 128;
    int r = i >> 2, c4 = i & 3;
    g2l_b128(Ag + (unsigned)r * K + c4 * 4, &As[r][c4 * 4]);
  }
  // B tile 16x64 -> transposed scatter, b32 DMA, 8 per thread
#pragma unroll
  for (int j = 0; j < 8; ++j) {
    int i = tid + j * 128;
    int r = i >> 6, c = i & 63;
    g2l_b32(Bg + (unsigned)r * N + c, &Bst[c][r]);
  }
#else
  // Fallback: batch all global loads into registers, then all LDS stores.
  float at[8], bt[8];
#pragma unroll
  for (int j = 0; j < 8; ++j) {
    int i = tid + j * 128;
    at[j] = Ag[(unsigned)(i >> 4) * K + (i & 15)];
  }
#pragma unroll
  for (int j = 0; j < 8; ++j) {
    int i = tid + j * 128;
    bt[j] = Bg[(unsigned)(i >> 6) * N + (i & 63)];
  }
#pragma unroll
  for (int j = 0; j < 8; ++j) {
    int i = tid + j * 128;
    As[i >> 4][i & 15] = at[j];
  }
#pragma unroll
  for (int j = 0; j < 8; ++j) {
    int i = tid + j * 128;
    Bst[i & 63][i >> 6] = bt[j];
  }
#endif
}

__global__ __launch_bounds__(128)
void wmma_gemm_bias(const float* __restrict__ A, const float* __restrict__ Bm,
                    const float* __restrict__ bias, float* __restrict__ Cmat,
                    int M, int N, int K) {
  __shared__ float As[2][TM][APAD];   // [buf][row][k]
  __shared__ float Bst[2][TN][BPAD];  // [buf][col][k] (transposed B tile)

  const int bm = blockIdx.y * TM;
  const int bn = blockIdx.x * TN;
  const int tid = threadIdx.x;
  const int wave = tid >> 5;
  const int lane = tid & 31;
  const int g = lane >> 4;   // half-wave group (selects K pair in A/B frags)
  const int ln = lane & 15;  // M index for A-frag / N index for B,C,D frags

  const float* Arow = A + (size_t)bm * K;   // 64-bit math once
  const float* Brow = Bm + bn;

  v8f acc0 = {}, acc1 = {}, acc2 = {}, acc3 = {};

  // Prologue: stage first K-tile
  stage_gemm_tile(As[0], Bst[0], Arow, Brow, (unsigned)K, (unsigned)N, tid);
  stage_wait();
  __syncthreads();

  int p = 0;
  const int am = wave * 16 + ln;  // A-frag row within block tile
  for (int k0 = 0; k0 < K; k0 += TK) {
    const int nk = k0 + TK;
    if (nk < K)  // stream next tile into the other buffer while computing
      stage_gemm_tile(As[p ^ 1], Bst[p ^ 1], Arow + nk, Brow + (size_t)nk * N,
                      (unsigned)K, (unsigned)N, tid);

    // Preload all fragments (aligned b64 LDS loads), then dense WMMA burst.
    v2f a[4], bf[4][4];
#pragma unroll
    for (int kq = 0; kq < 4; ++kq) {
      a[kq] = *reinterpret_cast<const v2f*>(&As[p][am][kq * 4 + 2 * g]);
#pragma unroll
      for (int t = 0; t < 4; ++t)
        bf[kq][t] = *reinterpret_cast<const v2f*>(&Bst[p][t * 16 + ln][kq * 4 + 2 * g]);
    }
#pragma unroll
    for (int kq = 0; kq < 4; ++kq) {
      acc0 = __builtin_amdgcn_wmma_f32_16x16x4_f32(false, a[kq], false, bf[kq][0], (short)0, acc0, false, false);
      acc1 = __builtin_amdgcn_wmma_f32_16x16x4_f32(false, a[kq], false, bf[kq][1], (short)0, acc1, false, false);
      acc2 = __builtin_amdgcn_wmma_f32_16x16x4_f32(false, a[kq], false, bf[kq][2], (short)0, acc2, false, false);
      acc3 = __builtin_amdgcn_wmma_f32_16x16x4_f32(false, a[kq], false, bf[kq][3], (short)0, acc3, false, false);
    }
    stage_wait();     // next-tile DMA complete (async path)
    __syncthreads();  // publish staged tile; retire LDS reads of buffer p
    p ^= 1;
  }

  // Epilogue: D layout -> global, fused bias. Row of VGPR r = r + 8*g.
  v8f accs[4] = {acc0, acc1, acc2, acc3};
#pragma unroll
  for (int t = 0; t < 4; ++t) {
    const int gcol = bn + t * 16 + ln;
    const float bv = bias[gcol];
#pragma unroll
    for (int r = 0; r < 8; ++r) {
      const unsigned grow = bm + wave * 16 + r + 8 * g;
      Cmat[grow * (unsigned)N + gcol] = accs[t][r] + bv;
    }
  }
}

// ---------------------------------------------------------------------------
// Flash-style causal attention, fp32 WMMA.
// Grid: (B*H, T/16). Block = 1 wave (32 threads), 16 query rows per wave.
// K tile staged as Ks[key][d] (b128 async; b64 frags along d); V tile staged
// transposed as Vst[d][key] (b32 async scatter; b64 frags along key); P
// round-trips through pbuf to convert the C/D layout into the A layout.
// qkv layout: [B, T, 3C] (q at col h*64, k at C + h*64, v at 2C + h*64)
// ---------------------------------------------------------------------------
__global__ __launch_bounds__(32)
void attn_kernel(const float* __restrict__ qkv, float* __restrict__ out) {
  const int bh = blockIdx.x;
  const int b = bh / H_;
  const int h = bh % H_;
  const int qtile = blockIdx.y;
  const int q_base = qtile * 16;
  const int lane = threadIdx.x;
  const int g = lane >> 4;
  const int ln = lane & 15;

  __shared__ float Ks[16][68];   // [key][d]   ; pad 68 -> 4n banks, 272B rows (16B aligned)
  __shared__ float Vst[64][20];  // [d][key]   ; pad 20, conflict-free
  __shared__ float pbuf[16][20]; // [row][key] ; pad 20, conflict-free

  const float scale = 0.125f;  // 1/sqrt(64)
  const float* qp = qkv + (size_t)b * T_ * C3_ + (size_t)h * D_;
  const float* kp = qp + C_;
  const float* vp = qp + 2 * C_;

  // Q fragments (A layout): 16 K-steps of 4 over head dim 64.
  v2f qf[16];
  const unsigned qoff = (unsigned)(q_base + ln) * C3_;
#pragma unroll
  for (int kb = 0; kb < D_; kb += 4) {
    qf[kb >> 2] = *reinterpret_cast<const v2f*>(&qp[qoff + kb + 2 * g]);
  }

  v8f o0 = {}, o1 = {}, o2 = {}, o3 = {};
  float mrun[8], lrun[8];
#pragma unroll
  for (int r = 0; r < 8; ++r) { mrun[r] = -INFINITY; lrun[r] = 0.0f; }

  for (int jt = 0; jt <= qtile; ++jt) {  // causal: skip fully-masked key tiles
    const int kbase = jt * 16;

    // ---- Stage K tile and transposed V tile ----
#if HAVE_ASYNC_G2L
#pragma unroll
    for (int j = 0; j < 8; ++j) {   // Ks: 256 16B chunks over 32 lanes
      int i = lane + j * 32;
      int r = i >> 4, c4 = i & 15;
      g2l_b128(kp + (unsigned)(kbase + r) * C3_ + c4 * 4, &Ks[r][c4 * 4]);
    }
#pragma unroll
    for (int j = 0; j < 32; ++j) {  // Vst: b32 transposed scatter
      int i = lane + j * 32;
      int k = i >> 6, c = i & 63;
      g2l_b32(vp + (unsigned)(kbase + k) * C3_ + c, &Vst[c][k]);
    }
    stage_wait();
#else
#pragma unroll
    for (int j = 0; j < 8; ++j) {
      int i = lane + j * 32;
      int r = i >> 4, c4 = i & 15;
      *reinterpret_cast<float4*>(&Ks[r][c4 * 4]) =
          *reinterpret_cast<const float4*>(&kp[(unsigned)(kbase + r) * C3_ + c4 * 4]);
    }
#pragma unroll
    for (int grp = 0; grp < 4; ++grp) {  // batch 8 loads, then 8 stores
      float t[8];
#pragma unroll
      for (int j = 0; j < 8; ++j) {
        int i = lane + (grp * 8 + j) * 32;
        t[j] = vp[(unsigned)(kbase + (i >> 6)) * C3_ + (i & 63)];
      }
#pragma unroll
      for (int j = 0; j < 8; ++j) {
        int i = lane + (grp * 8 + j) * 32;
        Vst[i & 63][i >> 6] = t[j];
      }
    }
    if (jt < qtile) {  // prefetch next key tile (global_prefetch_b8)
      const int nb = kbase + 16;
      __builtin_prefetch(&kp[(unsigned)(nb + (lane >> 1)) * C3_ + (lane & 1) * 32], 0, 0);
      __builtin_prefetch(&vp[(unsigned)(nb + (lane >> 1)) * C3_ + (lane & 1) * 32], 0, 0);
    }
#endif
    __syncthreads();

    // ---- S = Q * K^T (16x16): B-frag lane = key col, b64 along d ----
    v8f s = {};
#pragma unroll
    for (int kb = 0; kb < D_; kb += 4) {
      v2f bk = *reinterpret_cast<const v2f*>(&Ks[ln][kb + 2 * g]);
      s = __builtin_amdgcn_wmma_f32_16x16x4_f32(false, qf[kb >> 2], false, bk, (short)0, s, false, false);
    }

    // ---- Online softmax (rows live in 16-lane half-groups per VGPR) ----
    const int keycol = kbase + ln;
    float pv[8], alpha[8];
#pragma unroll
    for (int r = 0; r < 8; ++r) {
      const int qrow = q_base + r + 8 * g;
      float sv = s[r] * scale + (keycol > qrow ? -1e9f : 0.0f);
      float mx = sv;
      mx = fmaxf(mx, __shfl_xor(mx, 8, 16));
      mx = fmaxf(mx, __shfl_xor(mx, 4, 16));
      mx = fmaxf(mx, __shfl_xor(mx, 2, 16));
      mx = fmaxf(mx, __shfl_xor(mx, 1, 16));
      const float mnew = fmaxf(mrun[r], mx);
      const float p = __expf(sv - mnew);
      float sum = p;
      sum += __shfl_xor(sum, 8, 16);
      sum += __shfl_xor(sum, 4, 16);
      sum += __shfl_xor(sum, 2, 16);
      sum += __shfl_xor(sum, 1, 16);
      const float al = __expf(mrun[r] - mnew);
      lrun[r] = lrun[r] * al + sum;
      mrun[r] = mnew;
      alpha[r] = al;
      pv[r] = p;
    }
#pragma unroll
    for (int r = 0; r < 8; ++r) {
      o0[r] *= alpha[r]; o1[r] *= alpha[r]; o2[r] *= alpha[r]; o3[r] *= alpha[r];
    }

    // ---- P: D layout -> LDS -> A layout ----
#pragma unroll
    for (int r = 0; r < 8; ++r) pbuf[r + 8 * g][ln] = pv[r];
    __syncthreads();

    // ---- O += P * V: preload frags, then dense WMMA burst ----
    v2f pa[4], bv[4][4];
#pragma unroll
    for (int kq = 0; kq < 4; ++kq) {
      pa[kq] = *reinterpret_cast<const v2f*>(&pbuf[ln][kq * 4 + 2 * g]);
#pragma unroll
      for (int t = 0; t < 4; ++t)
        bv[kq][t] = *reinterpret_cast<const v2f*>(&Vst[t * 16 + ln][kq * 4 + 2 * g]);
    }
#pragma unroll
    for (int kq = 0; kq < 4; ++kq) {
      o0 = __builtin_amdgcn_wmma_f32_16x16x4_f32(false, pa[kq], false, bv[kq][0], (short)0, o0, false, false);
      o1 = __builtin_amdgcn_wmma_f32_16x16x4_f32(false, pa[kq], false, bv[kq][1], (short)0, o1, false, false);
      o2 = __builtin_amdgcn_wmma_f32_16x16x4_f32(false, pa[kq], false, bv[kq][2], (short)0, o2, false, false);
      o3 = __builtin_amdgcn_wmma_f32_16x16x4_f32(false, pa[kq], false, bv[kq][3], (short)0, o3, false, false);
    }
    __syncthreads();
  }

  // Normalize and store to [B, T, C] with head column offset
  v8f os[4] = {o0, o1, o2, o3};
#pragma unroll
  for (int t = 0; t < 4; ++t) {
    const int col = h * D_ + t * 16 + ln;
#pragma unroll
    for (int r = 0; r < 8; ++r) {
      const unsigned qrow = q_base + r + 8 * g;
      out[((unsigned)b * T_ + qrow) * C_ + col] = os[t][r] / lrun[r];
    }
  }
}

// ---------------------------------------------------------------------------
// Launch: qkv GEMM -> attention -> output GEMM, all on `stream`.
// Workspace: [0, 48MB) qkv [B,T,3C] fp32; [48MB, 64MB) attn out [B,T,C] fp32.
// ---------------------------------------------------------------------------
extern "C" void kernel_launch(void* const* d_in, const int* in_sizes, int n_in,
                              void* d_out, int out_size, void* d_ws, size_t ws_size,
                              hipStream_t stream) {
  const float* x     = (const float*)d_in[0];
  // d_in[1] = additive causal mask; causality is applied analytically in-kernel.
  const float* W_in  = (const float*)d_in[2];
  const float* b_in  = (const float*)d_in[3];
  const float* W_out = (const float*)d_in[4];
  const float* b_out = (const float*)d_in[5];
  float* outp = (float*)d_out;

  float* qkv_ws  = (float*)d_ws;                                   // 4096*3072 fp32
  float* attn_ws = (float*)((char*)d_ws + (size_t)B_ * T_ * C3_ * sizeof(float));

  const int M = B_ * T_;  // 4096

  // 1) QKV projection: [4096,1024] x [1024,3072] + b_in
  {
    dim3 grid(C3_ / TN, M / TM);
    wmma_gemm_bias<<<grid, dim3(128), 0, stream>>>(x, W_in, b_in, qkv_ws, M, C3_, C_);
  }
  // 2) Causal attention per (b,h), 16 query rows per wave
  {
    dim3 grid(B_ * H_, T_ / 16);
    attn_kernel<<<grid, dim3(32), 0, stream>>>(qkv_ws, attn_ws);
  }
  // 3) Output projection: [4096,1024] x [1024,1024] + b_out
  {
    dim3 grid(C_ / TN, M / TM);
    wmma_gemm_bias<<<grid, dim3(128), 0, stream>>>(attn_ws, W_out, b_out, outp, M, C_, C_);
  }
}